// OTDDDistance_63367947485237
// MI455X (gfx1250) — compile-verified
//
#include <hip/hip_runtime.h>

typedef float v2f __attribute__((ext_vector_type(2)));
typedef float v4f __attribute__((ext_vector_type(4)));
typedef float v8f __attribute__((ext_vector_type(8)));

#define NSAMP 2048
#define KDIM  64
#define DDIM  512
#define NCLS  10
#define CPAD  16
#define REGP  0.1f
#define MAX_ITER 100

// ---------------------------------------------------------------------------
// K1: streaming mean over axis 1.  536 MB of reads -> this is the whole
// kernel's runtime (~23 us at 23.3 TB/s HBM).  b128 non-temporal loads
// (input is single-use and >> 192MB L2); means (8 MB total) written with
// default RT hint so they stay resident in L2 for K2.
// ---------------------------------------------------------------------------
__global__ __launch_bounds__(128) void
k_mean(const float* __restrict__ x, const float* __restrict__ y,
       float* __restrict__ mx, float* __restrict__ my) {
  const int n      = blockIdx.x;       // sample
  const int tensor = blockIdx.y;       // 0 = x, 1 = y
  const float* src = tensor ? y : x;
  float*       dst = tensor ? my : mx;
  const int d = threadIdx.x * 4;       // 128 threads * float4 = 512 cols
  const float* base = src + (size_t)n * KDIM * DDIM + d;
  v4f acc = {0.f, 0.f, 0.f, 0.f};
#pragma unroll 16
  for (int k = 0; k < KDIM; ++k) {
    v4f t = __builtin_nontemporal_load((const v4f*)(base + (size_t)k * DDIM));
    acc += t;
  }
  acc *= (1.0f / (float)KDIM);
  *(v4f*)(dst + (size_t)n * DDIM + d) = acc;
}

// ---------------------------------------------------------------------------
// K2: class sums  S[c,d] = sum_n onehot(lab[n]==c) * means[n,d]
// as a [16 x 2048] @ [2048 x 512] matmul on the matrix cores:
// V_WMMA_F32_16X16X4_F32, one wave per (tensor, 16-wide d-tile).
// A (16x4 f32) layout: lane L -> M = L%16; VGPR j -> K = j + 2*(L/16).
// B (4x16 f32) layout: lane L -> N = L%16; VGPR j -> K = j + 2*(L/16).
// C/D (16x16 f32):     VGPR r -> M = r + 8*(L/16); N = L%16.
// The one-hot A is synthesized in registers from the labels; classes are
// padded to 16 (rows 10..15 accumulate exact zeros).
// ---------------------------------------------------------------------------
__global__ __launch_bounds__(32) void
k_class_sums(const float* __restrict__ means_x, const float* __restrict__ means_y,
             const int* __restrict__ lab_x, const int* __restrict__ lab_y,
             float* __restrict__ sums_x, float* __restrict__ sums_y) {
  const int wid    = blockIdx.x;     // 0..63
  const int tensor = wid >> 5;
  const int dt     = wid & 31;       // d-tile index (32 tiles of 16)
  const float* means = tensor ? means_y : means_x;
  const int*   lab   = tensor ? lab_y  : lab_x;
  float*       sums  = tensor ? sums_y : sums_x;

  const int lane = threadIdx.x;      // wave32, all lanes active (EXEC all-1s)
  const int m    = lane & 15;        // M (class row) / N (d column)
  const int kh   = (lane >> 4) * 2;  // K sub-offset: 0 for lanes 0-15, 2 for 16-31
  const int d0   = dt * 16 + m;      // global d column for this lane

  v8f c = {0.f, 0.f, 0.f, 0.f, 0.f, 0.f, 0.f, 0.f};
  for (int n = 0; n < NSAMP; n += 4) {
    const int l0 = lab[n + kh];
    const int l1 = lab[n + kh + 1];
    v2f a, b;
    a.x = (l0 == m) ? 1.0f : 0.0f;
    a.y = (l1 == m) ? 1.0f : 0.0f;
    b.x = means[(size_t)(n + kh)     * DDIM + d0];
    b.y = means[(size_t)(n + kh + 1) * DDIM + d0];
    c = __builtin_amdgcn_wmma_f32_16x16x4_f32(
        /*neg_a=*/false, a, /*neg_b=*/false, b,
        /*c_mod=*/(short)0, c, /*reuse_a=*/false, /*reuse_b=*/false);
  }
  const int rowbase = (lane >> 4) * 8;
#pragma unroll
  for (int r = 0; r < 8; ++r)
    sums[(size_t)(rowbase + r) * DDIM + d0] = c[r];
}

// ---------------------------------------------------------------------------
// K3: counts, probs, 10x10 cost, 100-iteration Sinkhorn, scalar output.
// All trivial work; single workgroup.
// ---------------------------------------------------------------------------
__global__ __launch_bounds__(128) void
k_finish(const float* __restrict__ sums_x, const float* __restrict__ sums_y,
         const int* __restrict__ lab_x, const int* __restrict__ lab_y,
         float* __restrict__ out) {
  __shared__ int   cntx[CPAD], cnty[CPAD];
  __shared__ float cost[NCLS * NCLS];
  __shared__ float Kx[NCLS * NCLS];
  __shared__ float uu[NCLS], vv[NCLS], px[NCLS], py[NCLS];
  const int tid = threadIdx.x;

  if (tid < CPAD) { cntx[tid] = 0; cnty[tid] = 0; }
  __syncthreads();
  for (int i = tid; i < NSAMP; i += blockDim.x) {
    atomicAdd(&cntx[lab_x[i]], 1);
    atomicAdd(&cnty[lab_y[i]], 1);
  }
  __syncthreads();

  if (tid == 0) {
    float ex[NCLS], ey[NCLS], totx = 0.f, toty = 0.f;
    for (int cc = 0; cc < NCLS; ++cc) {
      ex[cc] = cntx[cc] > 0 ? (float)cntx[cc] : 1e-6f; totx += ex[cc];
      ey[cc] = cnty[cc] > 0 ? (float)cnty[cc] : 1e-6f; toty += ey[cc];
    }
    for (int cc = 0; cc < NCLS; ++cc) {
      px[cc] = ex[cc] / (totx + 1e-10f);
      py[cc] = ey[cc] / (toty + 1e-10f);
    }
  }
  __syncthreads();

  if (tid < NCLS * NCLS) {
    const int i = tid / NCLS, j = tid % NCLS;
    const float invx = 1.0f / fmaxf((float)cntx[i], 1.0f);
    const float invy = 1.0f / fmaxf((float)cnty[j], 1.0f);
    float acc = 0.f;
    for (int d = 0; d < DDIM; ++d) {
      const float diff = sums_x[i * DDIM + d] * invx - sums_y[j * DDIM + d] * invy;
      acc += diff * diff;
    }
    cost[tid] = acc;
    Kx[tid]   = expf(-acc / REGP);
  }
  __syncthreads();

  if (tid < NCLS) { uu[tid] = 1.0f / NCLS; vv[tid] = 1.0f / NCLS; }
  __syncthreads();
  for (int it = 0; it < MAX_ITER; ++it) {
    if (tid < NCLS) {
      float s = 0.f;
      for (int j = 0; j < NCLS; ++j) s += Kx[tid * NCLS + j] * vv[j];
      uu[tid] = px[tid] / (s + 1e-10f);
    }
    __syncthreads();
    if (tid < NCLS) {
      float s = 0.f;
      for (int i = 0; i < NCLS; ++i) s += Kx[i * NCLS + tid] * uu[i];
      vv[tid] = py[tid] / (s + 1e-10f);
    }
    __syncthreads();
  }
  if (tid == 0) {
    float ot = 0.f;
    for (int i = 0; i < NCLS; ++i)
      for (int j = 0; j < NCLS; ++j)
        ot += uu[i] * Kx[i * NCLS + j] * vv[j] * cost[i * NCLS + j];
    out[0] = ot;
  }
}

// ---------------------------------------------------------------------------
extern "C" void kernel_launch(void* const* d_in, const int* in_sizes, int n_in,
                              void* d_out, int out_size, void* d_ws, size_t ws_size,
                              hipStream_t stream) {
  const float* x  = (const float*)d_in[0];
  const float* y  = (const float*)d_in[1];
  const int*   lx = (const int*)d_in[2];
  const int*   ly = (const int*)d_in[3];

  float* ws = (float*)d_ws;
  float* mx = ws;                                   // [2048 x 512]
  float* my = mx + (size_t)NSAMP * DDIM;            // [2048 x 512]
  float* sx = my + (size_t)NSAMP * DDIM;            // [16 x 512]
  float* sy = sx + (size_t)CPAD * DDIM;             // [16 x 512]

  k_mean<<<dim3(NSAMP, 2), 128, 0, stream>>>(x, y, mx, my);
  k_class_sums<<<64, 32, 0, stream>>>(mx, my, lx, ly, sx, sy);
  k_finish<<<1, 128, 0, stream>>>(sx, sy, lx, ly, (float*)d_out);
}